// SpAdjDropEdge_2774548873856
// MI455X (gfx1250) — compile-verified
//
#include <hip/hip_runtime.h>
#include <cstdint>

// ---------------------------------------------------------------------------
// SpAdjDropEdge for MI455X (gfx1250).
//   out[0:E]  = vals * (floor(threefry_uniform(key=42) + keep) / keep)
//   out[E:3E] = idxs (bit-identical int32 copy)
// Memory-bound (~154 MB total -> ~6.6 us at 23.3 TB/s). Threefry2x32 is
// computed inline in VALU (no extra memory traffic); vals are moved with
// B128 transactions. The 102 MB idxs copy uses the CDNA5 async
// global<->LDS data mover (ASYNCcnt) streaming 16B chunks through LDS
// without a VGPR round trip.
// ---------------------------------------------------------------------------

__device__ __forceinline__ uint32_t rotl32(uint32_t v, int n) {
    return (v << n) | (v >> (32 - n));   // lowers to v_alignbit_b32
}

// Threefry-2x32, 20 rounds, matching JAX's threefry2x32 exactly.
__device__ __forceinline__ void threefry2x32(uint32_t k0, uint32_t k1,
                                             uint32_t x0, uint32_t x1,
                                             uint32_t& o0, uint32_t& o1) {
    const uint32_t ks0 = k0;
    const uint32_t ks1 = k1;
    const uint32_t ks2 = k0 ^ k1 ^ 0x1BD11BDAu;

    x0 += ks0; x1 += ks1;
    // rounds 1-4
    x0 += x1; x1 = rotl32(x1, 13); x1 ^= x0;
    x0 += x1; x1 = rotl32(x1, 15); x1 ^= x0;
    x0 += x1; x1 = rotl32(x1, 26); x1 ^= x0;
    x0 += x1; x1 = rotl32(x1,  6); x1 ^= x0;
    x0 += ks1; x1 += ks2 + 1u;
    // rounds 5-8
    x0 += x1; x1 = rotl32(x1, 17); x1 ^= x0;
    x0 += x1; x1 = rotl32(x1, 29); x1 ^= x0;
    x0 += x1; x1 = rotl32(x1, 16); x1 ^= x0;
    x0 += x1; x1 = rotl32(x1, 24); x1 ^= x0;
    x0 += ks2; x1 += ks0 + 2u;
    // rounds 9-12
    x0 += x1; x1 = rotl32(x1, 13); x1 ^= x0;
    x0 += x1; x1 = rotl32(x1, 15); x1 ^= x0;
    x0 += x1; x1 = rotl32(x1, 26); x1 ^= x0;
    x0 += x1; x1 = rotl32(x1,  6); x1 ^= x0;
    x0 += ks0; x1 += ks1 + 3u;
    // rounds 13-16
    x0 += x1; x1 = rotl32(x1, 17); x1 ^= x0;
    x0 += x1; x1 = rotl32(x1, 29); x1 ^= x0;
    x0 += x1; x1 = rotl32(x1, 16); x1 ^= x0;
    x0 += x1; x1 = rotl32(x1, 24); x1 ^= x0;
    x0 += ks1; x1 += ks2 + 4u;
    // rounds 17-20
    x0 += x1; x1 = rotl32(x1, 13); x1 ^= x0;
    x0 += x1; x1 = rotl32(x1, 15); x1 ^= x0;
    x0 += x1; x1 = rotl32(x1, 26); x1 ^= x0;
    x0 += x1; x1 = rotl32(x1,  6); x1 ^= x0;
    x0 += ks2; x1 += ks0 + 5u;

    o0 = x0; o1 = x1;
}

__device__ __forceinline__ float tf_uniform(uint32_t bits) {
    return __uint_as_float((bits >> 9) | 0x3F800000u) - 1.0f;
}

// Vectorized path: each thread handles 4 consecutive counter blocks
// (c0..c0+3, paired with c0+half..), moving vals with B128 transactions.
// Requires half % 4 == 0 (true for E = 6,400,000).
__global__ void __launch_bounds__(256)
drop_vals_kernel_v4(const float* __restrict__ vals,
                    const float* __restrict__ keepPtr,
                    float* __restrict__ out,
                    long long half) {
    const long long c0 =
        4LL * ((long long)blockIdx.x * blockDim.x + threadIdx.x);
    if (c0 >= half) return;

    const float keep = keepPtr[0];
    const float inv  = 1.0f / keep;

    const float4 vlo = *(const float4*)(vals + c0);
    const float4 vhi = *(const float4*)(vals + half + c0);

    float slo[4], shi[4];
#pragma unroll
    for (int k = 0; k < 4; ++k) {
        uint32_t b0, b1;
        threefry2x32(0u, 42u, (uint32_t)(c0 + k),
                     (uint32_t)(c0 + k + half), b0, b1);
        slo[k] = floorf(tf_uniform(b0) + keep) * inv;   // {0, 1/keep}
        shi[k] = floorf(tf_uniform(b1) + keep) * inv;
    }

    float4 olo, ohi;
    olo.x = vlo.x * slo[0]; olo.y = vlo.y * slo[1];
    olo.z = vlo.z * slo[2]; olo.w = vlo.w * slo[3];
    ohi.x = vhi.x * shi[0]; ohi.y = vhi.y * shi[1];
    ohi.z = vhi.z * shi[2]; ohi.w = vhi.w * shi[3];

    *(float4*)(out + c0)        = olo;
    *(float4*)(out + half + c0) = ohi;
}

// Scalar fallback (only used when half % 4 != 0).
__global__ void __launch_bounds__(256)
drop_vals_kernel_s(const float* __restrict__ vals,
                   const float* __restrict__ keepPtr,
                   float* __restrict__ out,
                   long long half) {
    const long long j = (long long)blockIdx.x * blockDim.x + threadIdx.x;
    if (j >= half) return;

    const float keep = keepPtr[0];
    const float inv  = 1.0f / keep;

    uint32_t b0, b1;
    threefry2x32(0u, 42u, (uint32_t)j, (uint32_t)(j + half), b0, b1);

    out[j]        = vals[j]        * (floorf(tf_uniform(b0) + keep) * inv);
    out[j + half] = vals[j + half] * (floorf(tf_uniform(b1) + keep) * inv);
}

// ---------------------------------------------------------------------------
// idxs copy via CDNA5 async global<->LDS path. Each thread owns
// CHUNKS_PER_THREAD private 16B LDS slots, so no workgroup barrier is needed:
//   issue async loads -> s_wait_asynccnt 0 -> issue async stores -> wait.
// ---------------------------------------------------------------------------
#define COPY_TPB 256
#define CHUNKS_PER_THREAD 4

__global__ void __launch_bounds__(COPY_TPB)
copy_idxs_async_kernel(const int4* __restrict__ src,
                       int4* __restrict__ dst,
                       long long nChunks,
                       const int* __restrict__ srcTail,
                       int* __restrict__ dstTail,
                       int tailCount) {
    __shared__ int4 stage[COPY_TPB * CHUNKS_PER_THREAD];
    const int t = threadIdx.x;
    const long long blockBase =
        (long long)blockIdx.x * (COPY_TPB * CHUNKS_PER_THREAD);

#pragma unroll
    for (int k = 0; k < CHUNKS_PER_THREAD; ++k) {
        const long long c = blockBase + (long long)k * COPY_TPB + t;
        if (c < nChunks) {
            // Workgroup-relative LDS byte offset: low 32 bits of the generic ptr.
            const uint32_t loff =
                (uint32_t)(uintptr_t)&stage[k * COPY_TPB + t];
            const uint64_t gaddr = (uint64_t)(uintptr_t)(src + c);
            asm volatile("global_load_async_to_lds_b128 %0, %1, off"
                         :
                         : "v"(loff), "v"(gaddr)
                         : "memory");
        }
    }
    asm volatile("s_wait_asynccnt 0" ::: "memory");

#pragma unroll
    for (int k = 0; k < CHUNKS_PER_THREAD; ++k) {
        const long long c = blockBase + (long long)k * COPY_TPB + t;
        if (c < nChunks) {
            const uint32_t loff =
                (uint32_t)(uintptr_t)&stage[k * COPY_TPB + t];
            const uint64_t gaddr = (uint64_t)(uintptr_t)(dst + c);
            asm volatile("global_store_async_from_lds_b128 %0, %1, off"
                         :
                         : "v"(gaddr), "v"(loff)
                         : "memory");
        }
    }
    asm volatile("s_wait_asynccnt 0" ::: "memory");

    // Scalar tail (n % 4 elements), handled by block 0.
    if (blockIdx.x == 0 && t < tailCount) {
        dstTail[t] = srcTail[t];
    }
}

extern "C" void kernel_launch(void* const* d_in, const int* in_sizes, int n_in,
                              void* d_out, int out_size, void* d_ws, size_t ws_size,
                              hipStream_t stream) {
    const float* vals    = (const float*)d_in[0];
    const int*   idxs    = (const int*)d_in[1];
    const float* keepPtr = (const float*)d_in[2];
    float*       out     = (float*)d_out;

    const long long E    = (long long)in_sizes[0];
    const long long half = E / 2;   // E = 6,400,000 (even)

    // --- new_vals = vals * (mask / keepRate) ---
    if ((half & 3) == 0) {
        const int tpb = 256;
        const long long nThreads = half / 4;
        const int blocks = (int)((nThreads + tpb - 1) / tpb);
        drop_vals_kernel_v4<<<blocks, tpb, 0, stream>>>(vals, keepPtr, out, half);
    } else {
        const int tpb = 256;
        const int blocks = (int)((half + tpb - 1) / tpb);
        drop_vals_kernel_s<<<blocks, tpb, 0, stream>>>(vals, keepPtr, out, half);
    }

    // --- idxs bit-copy into out[E : 3E] ---
    {
        const long long n2 = 2 * E;           // #int32 to copy
        int* dstIdx = (int*)(out + E);        // E % 4 == 0 -> 16B aligned

        const long long nChunks = n2 / 4;     // 16B chunks
        const int tail = (int)(n2 % 4);

        const long long chunksPerBlock = (long long)COPY_TPB * CHUNKS_PER_THREAD;
        int cblocks = (int)((nChunks + chunksPerBlock - 1) / chunksPerBlock);
        if (cblocks == 0) cblocks = 1;

        copy_idxs_async_kernel<<<cblocks, COPY_TPB, 0, stream>>>(
            (const int4*)idxs, (int4*)dstIdx, nChunks,
            idxs + nChunks * 4, dstIdx + nChunks * 4, tail);
    }
}